// GCN_61856118997474
// MI455X (gfx1250) — compile-verified
//
#include <hip/hip_runtime.h>

#define N_NODES   100000
#define N_EDGES   3200000
#define IN_FEATS  512
#define HIDDEN    16
#define OUT_FEATS 128
#define N_CLASSES 3
#define N_TILES   (N_NODES / 16)   // 6250, exact

typedef __attribute__((ext_vector_type(2))) float v2f;
typedef __attribute__((ext_vector_type(8))) float v8f;

// ---------------------------------------------------------------- degrees
__global__ void gcn_deg_kernel(const int* __restrict__ src, const int* __restrict__ dst,
                               float* __restrict__ deg_out, float* __restrict__ deg_in) {
    int e = blockIdx.x * blockDim.x + threadIdx.x;
    if (e < N_EDGES) {
        atomicAdd(&deg_out[src[e]], 1.0f);
        atomicAdd(&deg_in[dst[e]], 1.0f);
    }
}

// in-place: deg -> rsqrt(max(deg,1))
__global__ void gcn_norm_kernel(float* __restrict__ deg_out, float* __restrict__ deg_in) {
    int i = blockIdx.x * blockDim.x + threadIdx.x;
    if (i < N_NODES) {
        deg_out[i] = rsqrtf(fmaxf(deg_out[i], 1.0f));
        deg_in[i]  = rsqrtf(fmaxf(deg_in[i], 1.0f));
    }
}

// ------------------------------------------------- GEMM1: h1 = (feat*ns) @ W1
// One wave handles a 16-node x 16-hidden tile, K=512 in steps of 4 (f32 WMMA).
// W1 (32 KB) is staged in LDS once per block; B-fragments come from ds loads.
__global__ void gcn_gemm1_wmma(const float* __restrict__ feat,
                               const float* __restrict__ norm_src,
                               const float* __restrict__ W1,
                               float* __restrict__ h1) {
    __shared__ float w1s[IN_FEATS * HIDDEN];           // 32 KB of 320 KB/WGP
    for (int i = threadIdx.x; i < IN_FEATS * HIDDEN; i += blockDim.x)
        w1s[i] = W1[i];
    __syncthreads();

    int wave = (blockIdx.x * blockDim.x + threadIdx.x) >> 5;
    if (wave >= N_TILES) return;          // wave-uniform exit; active waves keep EXEC=all-1s
    int lane = threadIdx.x & 31;
    int hh   = lane >> 4;                 // half-wave select: K pairs {0,1} vs {2,3}
    int lr   = lane & 15;                 // M for A-frag, N for B/D-frags
    int row0 = wave * 16;
    int node = row0 + lr;
    float ns = norm_src[node];
    const float* frow = feat + (size_t)node * IN_FEATS;

    v8f acc = {};
    for (int k = 0; k < IN_FEATS; k += 4) {
        int ka = k + hh * 2;
        v2f a; a.x = frow[ka] * ns;          a.y = frow[ka + 1] * ns;
        v2f b; b.x = w1s[ka * HIDDEN + lr];  b.y = w1s[(ka + 1) * HIDDEN + lr];
        acc = __builtin_amdgcn_wmma_f32_16x16x4_f32(false, a, false, b,
                                                    (short)0, acc, false, false);
    }
#pragma unroll
    for (int v = 0; v < 8; ++v) {
        int m = v + hh * 8;                  // D layout: vgpr v -> M=v (lanes<16) / v+8
        h1[(size_t)(row0 + m) * HIDDEN + lr] = acc[v];
    }
}

// --------------------------------------------------- edge scatter, 16 feats
// 4 threads per edge, float4 per thread. N_EDGES*4 divisible by 256 (no tail).
__global__ void gcn_scatter16(const int* __restrict__ src, const int* __restrict__ dst,
                              const float* __restrict__ h, float* __restrict__ agg) {
    int idx = blockIdx.x * blockDim.x + threadIdx.x;   // < N_EDGES*4 = 12.8M
    int e = idx >> 2;
    int q = (idx & 3) * 4;
    int s = src[e], d = dst[e];
    const float4 v = *(const float4*)(h + s * HIDDEN + q);
    float* a = agg + d * HIDDEN + q;
    atomicAdd(a + 0, v.x); atomicAdd(a + 1, v.y);
    atomicAdd(a + 2, v.z); atomicAdd(a + 3, v.w);
}

// x1 = relu(agg1 * norm_dst + b1), in place, float4-wide
__global__ void gcn_relu_bias(float4* __restrict__ agg1, const float* __restrict__ norm_dst,
                              const float* __restrict__ b1) {
    int idx = blockIdx.x * blockDim.x + threadIdx.x;   // < N_NODES*4
    if (idx < N_NODES * (HIDDEN / 4)) {
        int node = idx >> 2;
        int q    = (idx & 3) * 4;
        float nd = norm_dst[node];
        float4 v = agg1[idx];
        v.x = fmaxf(v.x * nd + b1[q + 0], 0.0f);
        v.y = fmaxf(v.y * nd + b1[q + 1], 0.0f);
        v.z = fmaxf(v.z * nd + b1[q + 2], 0.0f);
        v.w = fmaxf(v.w * nd + b1[q + 3], 0.0f);
        agg1[idx] = v;
    }
}

// ------------------------------------------------- GEMM2: h2 = (x1*ns) @ W2
// One wave: 16 nodes x 128 outputs = 8 N-tiles, K=16 => 4 WMMAs per tile.
// W2 (8 KB) staged in LDS.
__global__ void gcn_gemm2_wmma(const float* __restrict__ x1,
                               const float* __restrict__ norm_src,
                               const float* __restrict__ W2,
                               float* __restrict__ h2) {
    __shared__ float w2s[HIDDEN * OUT_FEATS];          // 8 KB
    for (int i = threadIdx.x; i < HIDDEN * OUT_FEATS; i += blockDim.x)
        w2s[i] = W2[i];
    __syncthreads();

    int wave = (blockIdx.x * blockDim.x + threadIdx.x) >> 5;
    if (wave >= N_TILES) return;
    int lane = threadIdx.x & 31;
    int hh   = lane >> 4;
    int lr   = lane & 15;
    int row0 = wave * 16;
    int node = row0 + lr;
    float ns = norm_src[node];

    v2f afrag[4];
#pragma unroll
    for (int s = 0; s < 4; ++s) {
        int ka = s * 4 + hh * 2;
        afrag[s].x = x1[node * HIDDEN + ka] * ns;
        afrag[s].y = x1[node * HIDDEN + ka + 1] * ns;
    }

    for (int nt = 0; nt < OUT_FEATS / 16; ++nt) {
        int n = nt * 16 + lr;
        v8f acc = {};
#pragma unroll
        for (int s = 0; s < 4; ++s) {
            int ka = s * 4 + hh * 2;
            v2f b; b.x = w2s[ka * OUT_FEATS + n]; b.y = w2s[(ka + 1) * OUT_FEATS + n];
            acc = __builtin_amdgcn_wmma_f32_16x16x4_f32(false, afrag[s], false, b,
                                                        (short)0, acc, false, false);
        }
#pragma unroll
        for (int v = 0; v < 8; ++v) {
            int m = v + hh * 8;
            h2[(size_t)(row0 + m) * OUT_FEATS + nt * 16 + lr] = acc[v];
        }
    }
}

// --------------------------------------------------- edge scatter, 128 feats
// One wave per edge: lane moves float4. N_EDGES*32 divisible by 256 (no tail);
// edge index is wave-uniform -> readfirstlane keeps address math scalar.
__global__ void gcn_scatter128(const int* __restrict__ src, const int* __restrict__ dst,
                               const float* __restrict__ h, float* __restrict__ agg) {
    int e    = (blockIdx.x * blockDim.x + threadIdx.x) >> 5;  // < N_EDGES
    int lane = threadIdx.x & 31;
    int s = __builtin_amdgcn_readfirstlane(src[e]);
    int d = __builtin_amdgcn_readfirstlane(dst[e]);
    const float4 v = *(const float4*)(h + (size_t)s * OUT_FEATS + lane * 4);
    float* a = agg + (size_t)d * OUT_FEATS + lane * 4;
    atomicAdd(a + 0, v.x); atomicAdd(a + 1, v.y);
    atomicAdd(a + 2, v.z); atomicAdd(a + 3, v.w);
}

// out = (agg2 * norm_dst + b2) @ Wfc + bfc
__global__ void gcn_fc(const float* __restrict__ agg2, const float* __restrict__ norm_dst,
                       const float* __restrict__ b2, const float* __restrict__ Wfc,
                       const float* __restrict__ bfc, float* __restrict__ out) {
    int idx = blockIdx.x * blockDim.x + threadIdx.x;
    if (idx < N_NODES * N_CLASSES) {
        int node = idx / N_CLASSES;
        int c    = idx % N_CLASSES;
        float nd = norm_dst[node];
        const float* row = agg2 + (size_t)node * OUT_FEATS;
        float s = bfc[c];
#pragma unroll 4
        for (int f = 0; f < OUT_FEATS; ++f)
            s += (row[f] * nd + b2[f]) * Wfc[f * N_CLASSES + c];
        out[idx] = s;
    }
}

extern "C" void kernel_launch(void* const* d_in, const int* in_sizes, int n_in,
                              void* d_out, int out_size, void* d_ws, size_t ws_size,
                              hipStream_t stream) {
    const float* feat = (const float*)d_in[0];
    const int*   src  = (const int*)d_in[1];
    const int*   dst  = (const int*)d_in[2];
    const float* W1   = (const float*)d_in[3];
    const float* b1   = (const float*)d_in[4];
    const float* W2   = (const float*)d_in[5];
    const float* b2   = (const float*)d_in[6];
    const float* Wfc  = (const float*)d_in[7];
    const float* bfc  = (const float*)d_in[8];
    float* out = (float*)d_out;

    // workspace layout (floats): deg/norm_src | deg/norm_dst | h1 | agg1(=x1) | h2 | agg2
    float* ws       = (float*)d_ws;
    float* deg_out  = ws;                                   // N       (-> norm_src)
    float* deg_in   = deg_out + N_NODES;                    // N       (-> norm_dst)
    float* h1       = deg_in + N_NODES;                     // N*16
    float* agg1     = h1 + (size_t)N_NODES * HIDDEN;        // N*16    (-> x1 in place)
    float* h2       = agg1 + (size_t)N_NODES * HIDDEN;      // N*128
    float* agg2     = h2 + (size_t)N_NODES * OUT_FEATS;     // N*128

    hipMemsetAsync(deg_out, 0, sizeof(float) * (size_t)2 * N_NODES, stream);
    hipMemsetAsync(agg1,    0, sizeof(float) * (size_t)N_NODES * HIDDEN, stream);
    hipMemsetAsync(agg2,    0, sizeof(float) * (size_t)N_NODES * OUT_FEATS, stream);

    gcn_deg_kernel <<<(N_EDGES + 255) / 256, 256, 0, stream>>>(src, dst, deg_out, deg_in);
    gcn_norm_kernel<<<(N_NODES + 255) / 256, 256, 0, stream>>>(deg_out, deg_in);
    // deg_out now holds norm_src, deg_in holds norm_dst

    gcn_gemm1_wmma<<<(N_TILES + 7) / 8, 256, 0, stream>>>(feat, deg_out, W1, h1);

    gcn_scatter16<<<(N_EDGES * 4) / 256, 256, 0, stream>>>(src, dst, h1, agg1);

    gcn_relu_bias<<<(N_NODES * (HIDDEN / 4) + 255) / 256, 256, 0, stream>>>(
        (float4*)agg1, deg_in, b1);

    gcn_gemm2_wmma<<<(N_TILES + 7) / 8, 256, 0, stream>>>(agg1, deg_out, W2, h2);

    gcn_scatter128<<<(unsigned)(((long long)N_EDGES * 32) / 256), 256, 0, stream>>>(
        src, dst, h2, agg2);

    gcn_fc<<<(N_NODES * N_CLASSES + 255) / 256, 256, 0, stream>>>(agg2, deg_in, b2, Wfc, bfc, out);
}